// Model_25056839205188
// MI455X (gfx1250) — compile-verified
//
#include <hip/hip_runtime.h>
#include <math.h>

// MI455X / gfx1250 fused RMSNorm + projection + Sinkhorn kernel.
// Single pass over x (256 MB @ 23.3 TB/s => ~11us floor): WMMA f32 16x16x4
// carries the 24-col projection (25 FMA/loaded float is beyond VALU peak),
// VALU carries sum-of-squares, Sinkhorn epilogue runs in registers.
//
// m = (scale * x / rms) @ phi  =  (x @ (diag(scale) phi)) / rms
// so scale is folded into phi' during LDS staging and x is read exactly once.

typedef __attribute__((ext_vector_type(2))) float v2f;
typedef __attribute__((ext_vector_type(8))) float v8f;

#define KDIM        4096            // STREAM
#define KHALF       2048            // per-wave K split (2 waves reduce per tile)
#define NOUT        24              // n*n + 2n
#define NPAD        32              // two 16-wide WMMA N-tiles
#define KC          128             // K chunk staged in LDS per half
#define KP          (KC / 2)        // k-pairs per chunk (float2 interleaved)
#define LDS_S2      48              // float2 row stride; 48 ≡ 16 (mod 32) ->
                                    // rows p, p+1 hit disjoint bank halves
#define WAVES       8
#define BLOCK       256             // 8 x wave32
#define ROWS_PER_WG 64              // 4 row tiles x 16 rows
#define NUM_WG      256             // 16384 rows total -> 2048 waves
#define H_PRE_OFF   0
#define H_POST_OFF  65536           // 16384*4
#define H_RES_OFF   131072          // 2*16384*4

__global__ __launch_bounds__(BLOCK)
void fused_rms_proj_sinkhorn(const float* __restrict__ x,
                             const float* __restrict__ scale,
                             const float* __restrict__ phi,
                             const float* __restrict__ bias,
                             const float* __restrict__ a_pre_p,
                             const float* __restrict__ a_post_p,
                             const float* __restrict__ a_res_p,
                             float* __restrict__ out)
{
    // phi' chunks, k-pair interleaved: lds_phi2[half][p][col] = {phi'[k],phi'[k+1]}
    __shared__ v2f   lds_phi2[2 * KP * LDS_S2];        // 48 KB
    __shared__ float dmat[WAVES][16][NPAD];            // 16 KB: per-wave D tiles
    __shared__ float ssq_lds[BLOCK];                   // 1 KB

    const int t    = threadIdx.x;
    const int wave = t >> 5;
    const int lane = t & 31;
    const int wg   = blockIdx.x;

    const int rowtile = wave & 3;   // which 16-row tile
    const int ksplit  = wave >> 2;  // which K half
    const int n_idx   = lane & 15;  // M index for A / N index for B
    const int khalf   = lane >> 4;  // 0: K=0,1  1: K=2,3 (ISA 16x4 fp32 A layout)

    const int row0 = wg * ROWS_PER_WG + rowtile * 16;
    // lane's streaming pointer into its row of x (8B aligned)
    const float* xrow = x + (size_t)(row0 + n_idx) * KDIM
                          + (size_t)ksplit * KHALF + 2 * khalf;

    v8f c0 = {};   // D cols 0..15
    v8f c1 = {};   // D cols 16..31
    float ssq0 = 0.0f, ssq1 = 0.0f;

    for (int kc = 0; kc < KHALF; kc += KC) {
        // ---- stage phi' = scale ⊙ phi for BOTH K-half chunks, pair-interleaved ----
        for (int idx = t; idx < 2 * KP * NPAD; idx += BLOCK) {
            const int e    = idx & 31;
            const int p    = (idx >> 5) & (KP - 1);
            const int half = idx >> 11;            // 2*KP*NPAD = 4096 -> bit 11
            const int k0   = half * KHALF + kc + 2 * p;
            v2f v = {};
            if (e < NOUT) {
                v.x = scale[k0]     * phi[(size_t)k0       * NOUT + e];
                v.y = scale[k0 + 1] * phi[(size_t)(k0 + 1) * NOUT + e];
            }
            lds_phi2[(half * KP + p) * LDS_S2 + e] = v;
        }
        __syncthreads();

        const float* xp = xrow + kc;
        if (kc + KC < KHALF)
            __builtin_prefetch(xp + KC, 0, 1);     // global_prefetch next chunk

        const v2f* bbase = &lds_phi2[(ksplit * KP) * LDS_S2 + n_idx];

        #pragma unroll 4
        for (int kk = 0; kk < KC; kk += 4) {
            // A fragment: lanes 0-15 hold {k,k+1}; lanes 16-31 hold {k+2,k+3}
            const v2f a = *(const v2f*)(xp + kk);
            ssq0 = fmaf(a.x, a.x, ssq0);
            ssq1 = fmaf(a.y, a.y, ssq1);

            // B fragments: one ds_load_b64 each, pre-packed in LDS
            const v2f* bp = bbase + ((kk >> 1) + khalf) * LDS_S2;
            const v2f b0 = bp[0];
            const v2f b1 = bp[16];

            c0 = __builtin_amdgcn_wmma_f32_16x16x4_f32(
                     false, a, false, b0, (short)0, c0, false, false);
            c1 = __builtin_amdgcn_wmma_f32_16x16x4_f32(
                     false, a, false, b1, (short)0, c1, false, false);
        }
        __syncthreads();
    }

    // ---- dump D tiles + sum-of-squares partials to LDS ----
    #pragma unroll
    for (int i = 0; i < 8; ++i) {
        const int m = i + 8 * khalf;               // C/D layout: VGPR i -> M row
        dmat[wave][m][n_idx]      = c0[i];
        dmat[wave][m][n_idx + 16] = c1[i];
    }
    ssq_lds[t] = ssq0 + ssq1;
    __syncthreads();

    // ---- epilogue: one thread per row (64 of 256) ----
    if (t < ROWS_PER_WG) {
        const int rt  = t >> 4;                    // row tile 0..3
        const int r   = t & 15;
        const int row = wg * ROWS_PER_WG + t;

        // reduce the two K-half waves (rt and rt+4)
        const float s2 = ssq_lds[rt * 32 + r]       + ssq_lds[rt * 32 + r + 16]
                       + ssq_lds[(rt + 4) * 32 + r] + ssq_lds[(rt + 4) * 32 + r + 16];
        const float rinv = rsqrtf(s2 * (1.0f / KDIM) + 1e-20f);

        const float alpha_pre  = a_pre_p[0];
        const float alpha_post = a_post_p[0];
        const float alpha_res  = a_res_p[0];

        float m24[NOUT];
        #pragma unroll
        for (int e = 0; e < NOUT; ++e)
            m24[e] = (dmat[rt][r][e] + dmat[rt + 4][r][e]) * rinv + bias[e];

        float hp[4], hq[4], M[16];
        #pragma unroll
        for (int i = 0; i < 4; ++i) {
            hp[i] = 1.0f / (1.0f + __expf(-alpha_pre  * m24[i]));
            hq[i] = 2.0f / (1.0f + __expf(-alpha_post * m24[4 + i]));
        }
        #pragma unroll
        for (int i = 0; i < 16; ++i)
            M[i] = __expf(alpha_res * m24[8 + i]);

        // 20 Sinkhorn-Knopp iterations, fully in registers
        for (int it = 0; it < 20; ++it) {
            #pragma unroll
            for (int i = 0; i < 4; ++i) {
                const float inv = 1.0f /
                    (M[i*4+0] + M[i*4+1] + M[i*4+2] + M[i*4+3] + 1e-12f);
                M[i*4+0] *= inv; M[i*4+1] *= inv;
                M[i*4+2] *= inv; M[i*4+3] *= inv;
            }
            #pragma unroll
            for (int j = 0; j < 4; ++j) {
                const float inv = 1.0f /
                    (M[j] + M[4+j] + M[8+j] + M[12+j] + 1e-12f);
                M[j]    *= inv; M[4+j]  *= inv;
                M[8+j]  *= inv; M[12+j] *= inv;
            }
        }

        // outputs: h_pre | h_post | h_res concatenated flat
        *(float4*)(out + H_PRE_OFF  + (size_t)row * 4) =
            make_float4(hp[0], hp[1], hp[2], hp[3]);
        *(float4*)(out + H_POST_OFF + (size_t)row * 4) =
            make_float4(hq[0], hq[1], hq[2], hq[3]);
        float4* o_res = (float4*)(out + H_RES_OFF + (size_t)row * 16);
        #pragma unroll
        for (int i = 0; i < 4; ++i)
            o_res[i] = make_float4(M[i*4+0], M[i*4+1], M[i*4+2], M[i*4+3]);
    }
}

extern "C" void kernel_launch(void* const* d_in, const int* in_sizes, int n_in,
                              void* d_out, int out_size, void* d_ws, size_t ws_size,
                              hipStream_t stream)
{
    const float* x      = (const float*)d_in[0];   // [8,2048,4096]
    const float* scale  = (const float*)d_in[1];   // [4096]
    const float* phi    = (const float*)d_in[2];   // [4096,24]
    const float* bias   = (const float*)d_in[3];   // [1,24]
    const float* a_pre  = (const float*)d_in[4];   // scalar
    const float* a_post = (const float*)d_in[5];   // scalar
    const float* a_res  = (const float*)d_in[6];   // scalar
    float* out = (float*)d_out;

    fused_rms_proj_sinkhorn<<<NUM_WG, BLOCK, 0, stream>>>(
        x, scale, phi, bias, a_pre, a_post, a_res, out);
}